// SAGE_6554120093875
// MI455X (gfx1250) — compile-verified
//
#include <hip/hip_runtime.h>

#define NN 100000
#define NE 3200000
#define INC 11
#define HID 64

typedef __attribute__((ext_vector_type(2))) float v2f;
typedef __attribute__((ext_vector_type(8))) float v8f;

// ---------------- workspace zeroing ----------------
__global__ void zero_ws_kernel(float* __restrict__ ws, int n) {
  int i = blockIdx.x * blockDim.x + threadIdx.x;
  if (i < n) ws[i] = 0.0f;
}

// ---------------- layer-1 scatter: degree + 11-ch neighbor sum ----------------
__global__ void scatter1_kernel(const float* __restrict__ x,
                                const long long* __restrict__ ei,
                                float* __restrict__ cnt,
                                float* __restrict__ agg1) {
  int e = blockIdx.x * blockDim.x + threadIdx.x;
  if (e >= NE) return;
  int src = (int)ei[e];
  int dst = (int)ei[NE + e];
  atomicAdd(cnt + dst, 1.0f);                 // global_atomic_add_f32, no return
  const float* xs = x + src * INC;
  float* ad = agg1 + dst * INC;
#pragma unroll
  for (int c = 0; c < INC; ++c) atomicAdd(ad + c, xs[c]);
}

// ---------------- layer-1 dense: h1 = relu([mean|x] @ [Wl1;Wr1]^T + bl1) ------
// fp32 WMMA 16x16x4, K padded 22 -> 24 (6 K-steps), N = 64 (4 tiles)
__global__ void __launch_bounds__(256) layer1_kernel(
    const float* __restrict__ x, const float* __restrict__ cnt,
    const float* __restrict__ agg1,
    const float* __restrict__ Wl1, const float* __restrict__ bl1,
    const float* __restrict__ Wr1, float* __restrict__ h1) {
  __shared__ float Bs[24 * 72];   // stride 72: 2 rows apart -> +16 banks, conflict-free
  __shared__ float bias[64];
  int tid = threadIdx.x;
  for (int i = tid; i < 24 * 64; i += 256) {
    int k = i >> 6, j = i & 63;
    float v = 0.0f;
    if (k < 11)      v = Wl1[j * 11 + k];
    else if (k < 22) v = Wr1[j * 11 + (k - 11)];
    Bs[k * 72 + j] = v;
  }
  if (tid < 64) bias[tid] = bl1[tid];
  __syncthreads();

  int lane = tid & 31;
  int wave = tid >> 5;
  int half = lane >> 4;          // selects K-pair {0,1} vs {2,3} in A/B fragments
  int l15  = lane & 15;
  int base = blockIdx.x * 128 + wave * 16;
  int node = base + l15;
  int nodeC = node < NN ? node : (NN - 1);
  float vmask = node < NN ? 1.0f : 0.0f;
  float inv = vmask / fmaxf(cnt[nodeC], 1.0f);

  v8f acc[4] = {};
#pragma unroll
  for (int kb = 0; kb < 6; ++kb) {
    int k0 = kb * 4 + half * 2;
    int k1 = k0 + 1;
    v2f a;
    a.x = (k0 < 11) ? agg1[nodeC * 11 + k0] * inv
        : ((k0 < 22) ? x[nodeC * 11 + (k0 - 11)] * vmask : 0.0f);
    a.y = (k1 < 11) ? agg1[nodeC * 11 + k1] * inv
        : ((k1 < 22) ? x[nodeC * 11 + (k1 - 11)] * vmask : 0.0f);
    const float* brow = Bs + k0 * 72 + l15;
#pragma unroll
    for (int nt = 0; nt < 4; ++nt) {
      v2f b;
      b.x = brow[nt * 16];
      b.y = brow[72 + nt * 16];
      acc[nt] = __builtin_amdgcn_wmma_f32_16x16x4_f32(
          false, a, false, b, (short)0, acc[nt], false, false);
    }
  }
#pragma unroll
  for (int nt = 0; nt < 4; ++nt) {
#pragma unroll
    for (int r = 0; r < 8; ++r) {
      int ns = base + half * 8 + r;      // D layout: VGPR r -> M=r / M=8+r
      if (ns < NN) {
        int ch = nt * 16 + l15;
        h1[ns * 64 + ch] = fmaxf(acc[nt][r] + bias[ch], 0.0f);
      }
    }
  }
}

// ---------------- layer-2 scatter: 64-ch neighbor sum, float4 chunks ---------
__global__ void scatter2_kernel(const float* __restrict__ h1,
                                const long long* __restrict__ ei,
                                float* __restrict__ agg2) {
  long long t = (long long)blockIdx.x * blockDim.x + threadIdx.x;
  if (t >= (long long)NE * 16) return;
  int e = (int)(t >> 4);
  int chunk = (int)(t & 15);
  int src = (int)ei[e];
  int dst = (int)ei[NE + e];
  const float4 v = *(const float4*)(h1 + src * 64 + chunk * 4);  // 256B/edge coalesced
  float* ad = agg2 + dst * 64 + chunk * 4;
  atomicAdd(ad + 0, v.x);
  atomicAdd(ad + 1, v.y);
  atomicAdd(ad + 2, v.z);
  atomicAdd(ad + 3, v.w);
}

// ---- layer-2 dense + fused head: out = relu([mean2|h1]@[Wl2;Wr2]^T+bl2)@Wlin^T+blin
__global__ void __launch_bounds__(256) layer2_kernel(
    const float* __restrict__ h1, const float* __restrict__ cnt,
    const float* __restrict__ agg2,
    const float* __restrict__ Wl2, const float* __restrict__ bl2,
    const float* __restrict__ Wr2,
    const float* __restrict__ Wlin, const float* __restrict__ blin,
    float* __restrict__ out) {
  __shared__ float Bs[128 * 72];   // 36 KB of the 320 KB WGP LDS
  __shared__ float bias[64];
  __shared__ float wl[64];
  int tid = threadIdx.x;
  for (int i = tid; i < 128 * 64; i += 256) {
    int k = i >> 6, j = i & 63;
    Bs[k * 72 + j] = (k < 64) ? Wl2[j * 64 + k] : Wr2[j * 64 + (k - 64)];
  }
  if (tid < 64) { bias[tid] = bl2[tid]; wl[tid] = Wlin[tid]; }
  __syncthreads();

  int lane = tid & 31;
  int wave = tid >> 5;
  int half = lane >> 4;
  int l15  = lane & 15;
  int base = blockIdx.x * 128 + wave * 16;
  int node = base + l15;
  int nodeC = node < NN ? node : (NN - 1);
  float vmask = node < NN ? 1.0f : 0.0f;
  float inv = vmask / fmaxf(cnt[nodeC], 1.0f);

  v8f acc[4] = {};
#pragma unroll
  for (int kb = 0; kb < 32; ++kb) {          // K = 128 -> 32 x (16x16x4)
    int k0 = kb * 4 + half * 2;
    v2f a;
    if (kb < 16) {                            // compile-time after unroll
      a.x = agg2[nodeC * 64 + k0] * inv;
      a.y = agg2[nodeC * 64 + k0 + 1] * inv;
    } else {
      a.x = h1[nodeC * 64 + (k0 - 64)] * vmask;
      a.y = h1[nodeC * 64 + (k0 - 63)] * vmask;
    }
    const float* brow = Bs + k0 * 72 + l15;
#pragma unroll
    for (int nt = 0; nt < 4; ++nt) {
      v2f b;
      b.x = brow[nt * 16];
      b.y = brow[72 + nt * 16];
      acc[nt] = __builtin_amdgcn_wmma_f32_16x16x4_f32(
          false, a, false, b, (short)0, acc[nt], false, false);
    }
  }

  // fused head: relu + bias, dot with Wlin, reduce across the 16-lane half
  float bl0 = blin[0];
#pragma unroll
  for (int r = 0; r < 8; ++r) {
    float p = 0.0f;
#pragma unroll
    for (int nt = 0; nt < 4; ++nt) {
      int ch = nt * 16 + l15;
      p += fmaxf(acc[nt][r] + bias[ch], 0.0f) * wl[ch];
    }
    p += __shfl_xor(p, 1);
    p += __shfl_xor(p, 2);
    p += __shfl_xor(p, 4);
    p += __shfl_xor(p, 8);    // offsets < 16 keep the two halves independent (wave32)
    int ns = base + half * 8 + r;
    if (l15 == 0 && ns < NN) out[ns] = p + bl0;
  }
}

extern "C" void kernel_launch(void* const* d_in, const int* in_sizes, int n_in,
                              void* d_out, int out_size, void* d_ws, size_t ws_size,
                              hipStream_t stream) {
  const float* x        = (const float*)d_in[0];
  const long long* ei   = (const long long*)d_in[1];   // int64 edge_index [2, E]
  const float* Wl1      = (const float*)d_in[2];
  const float* bl1      = (const float*)d_in[3];
  const float* Wr1      = (const float*)d_in[4];
  const float* Wl2      = (const float*)d_in[5];
  const float* bl2      = (const float*)d_in[6];
  const float* Wr2      = (const float*)d_in[7];
  const float* Wlin     = (const float*)d_in[8];
  const float* blin     = (const float*)d_in[9];
  float* out = (float*)d_out;

  float* ws   = (float*)d_ws;
  float* cnt  = ws;                 // [N]
  float* agg1 = ws + NN;            // [N, 11]
  float* h1   = ws + 12 * NN;       // [N, 64]
  float* agg2 = ws + 76 * NN;       // [N, 64]   total 140*N floats = 56 MB

  int nzero = 140 * NN;
  zero_ws_kernel<<<(nzero + 255) / 256, 256, 0, stream>>>(ws, nzero);
  scatter1_kernel<<<(NE + 255) / 256, 256, 0, stream>>>(x, ei, cnt, agg1);
  layer1_kernel<<<(NN + 127) / 128, 256, 0, stream>>>(x, cnt, agg1, Wl1, bl1, Wr1, h1);
  long long t2 = (long long)NE * 16;
  scatter2_kernel<<<(int)((t2 + 255) / 256), 256, 0, stream>>>(h1, ei, agg2);
  layer2_kernel<<<(NN + 127) / 128, 256, 0, stream>>>(h1, cnt, agg2, Wl2, bl2, Wr2,
                                                      Wlin, blin, out);
}